// Warp_32573031973405
// MI455X (gfx1250) — compile-verified
//
#include <hip/hip_runtime.h>

// 3-D trilinear warp (grid_sample, zeros padding, align_corners=True)
// B=2, C=1, H=W=D=128.
// Memory-bound gather: image (16MB) stays L2-resident (192MB L2); ddf/out are
// one-shot streams -> non-temporal hints + global_prefetch. Each thread now
// handles 4 consecutive d-voxels so all streaming traffic is b128
// (3x global_load_b128 + 1x global_store_b128 per thread).
// HBM floor: (48+16+16)MB / 23.3 TB/s ~= 3.4us.

typedef float f4v __attribute__((ext_vector_type(4)));

#define DIM   128
#define NPB   (DIM * DIM * DIM)       // voxels per batch = 2,097,152
#define PFD   16384                   // prefetch distance in floats (64 KB)

__global__ __launch_bounds__(256)
void warp3d_trilinear_x4(const float* __restrict__ image,
                         const float* __restrict__ ddf,
                         float* __restrict__ out,
                         int nthreads)
{
    const int tid = blockIdx.x * blockDim.x + threadIdx.x;
    if (tid >= nthreads) return;

    const int gid4 = tid << 2;               // first of 4 consecutive voxels
    const int b    = gid4 >> 21;             // total voxels = 2 * 2^21
    const int rem  = gid4 & (NPB - 1);
    const int h    = rem >> 14;
    const int w    = (rem >> 7) & (DIM - 1);
    const int d0   = rem & (DIM - 1);        // multiple of 4, row-aligned

    // ---- streaming ddf loads: 16B-aligned b128, non-temporal (read-once) ----
    const float* ddfb = ddf + (size_t)b * 3u * NPB + rem;
    const f4v dh = __builtin_nontemporal_load((const f4v*)(ddfb));            // H disp
    const f4v dw = __builtin_nontemporal_load((const f4v*)(ddfb + NPB));      // W disp
    const f4v dd = __builtin_nontemporal_load((const f4v*)(ddfb + 2 * NPB));  // D disp

    // gfx1250 global_prefetch on the three ddf streams, 64KB ahead (GL2)
    __builtin_prefetch(ddfb + PFD,           0, 1);
    __builtin_prefetch(ddfb + NPB + PFD,     0, 1);
    __builtin_prefetch(ddfb + 2 * NPB + PFD, 0, 1);

    const float* __restrict__ img = image + (size_t)b * NPB;
    const float hf = (float)h;
    const float wf = (float)w;

    f4v res;
#pragma unroll
    for (int k = 0; k < 4; ++k) {
        // ---- absolute voxel coordinates for sub-voxel k ----
        const float cx = hf + dh[k];
        const float cy = wf + dw[k];
        const float cz = (float)(d0 + k) + dd[k];

        const float fx = floorf(cx), fy = floorf(cy), fz = floorf(cz);
        const int   ix = (int)fx,    iy = (int)fy,    iz = (int)fz;
        const float tx = cx - fx,    ty = cy - fy,    tz = cz - fz;

        // ---- 8 trilinear taps, branchless zero-padding ----
        float acc = 0.0f;
#pragma unroll
        for (int dx = 0; dx < 2; ++dx) {
            const int   x  = ix + dx;
            const bool  vx = (unsigned)x < (unsigned)DIM;
            const int   xc = min(max(x, 0), DIM - 1);
            const float wx = dx ? tx : 1.0f - tx;
#pragma unroll
            for (int dy = 0; dy < 2; ++dy) {
                const int   y    = iy + dy;
                const bool  vxy  = vx && ((unsigned)y < (unsigned)DIM);
                const int   yc   = min(max(y, 0), DIM - 1);
                const float wxy  = wx * (dy ? ty : 1.0f - ty);
                const int   base = ((xc << 7) + yc) << 7;
#pragma unroll
                for (int dz = 0; dz < 2; ++dz) {
                    const int   z  = iz + dz;
                    const bool  v  = vxy && ((unsigned)z < (unsigned)DIM);
                    const int   zc = min(max(z, 0), DIM - 1);
                    const float wv = wxy * (dz ? tz : 1.0f - tz);
                    const float val = img[base + zc];     // L2-resident gather
                    acc = fmaf(val, v ? wv : 0.0f, acc);
                }
            }
        }
        res[k] = acc;
    }

    // one-shot output stream: b128 non-temporal store
    __builtin_nontemporal_store(res, (f4v*)(out + gid4));
}

extern "C" void kernel_launch(void* const* d_in, const int* in_sizes, int n_in,
                              void* d_out, int out_size, void* d_ws, size_t ws_size,
                              hipStream_t stream) {
    const float* image = (const float*)d_in[0];   // (2,1,128,128,128) f32
    const float* ddf   = (const float*)d_in[1];   // (2,3,128,128,128) f32
    float*       out   = (float*)d_out;           // (2,1,128,128,128) f32

    const int nthreads = out_size >> 2;           // 4 voxels per thread
    const int block    = 256;                     // 8 wave32s
    const int grid     = (nthreads + block - 1) / block;

    warp3d_trilinear_x4<<<grid, block, 0, stream>>>(image, ddf, out, nthreads);
}